// MalwareGNN_8718783610905
// MI455X (gfx1250) — compile-verified
//
#include <hip/hip_runtime.h>
#include <hip/hip_bf16.h>

typedef float v2f __attribute__((ext_vector_type(2)));
typedef float v8f __attribute__((ext_vector_type(8)));

static constexpr int HDIM = 128;        // hidden width (also input F)
static constexpr int NGRAPH = 64;

// ---------------------------------------------------------------------------
// zero fill
// ---------------------------------------------------------------------------
__global__ void zero_kernel(float* __restrict__ p, long long n) {
    long long i = (long long)blockIdx.x * blockDim.x + threadIdx.x;
    if (i < n) p[i] = 0.0f;
}

// ---------------------------------------------------------------------------
// degree: deg[dst] += 1 over all edges
// ---------------------------------------------------------------------------
__global__ void deg_kernel(const int* __restrict__ dst, float* __restrict__ deg, int E) {
    int e = blockIdx.x * blockDim.x + threadIdx.x;
    if (e < E) atomicAdd(&deg[dst[e]], 1.0f);
}

__global__ void dinv_kernel(const float* __restrict__ deg,
                            float* __restrict__ dinv, float* __restrict__ invdeg, int N) {
    int i = blockIdx.x * blockDim.x + threadIdx.x;
    if (i < N) {
        float d = deg[i] + 1.0f;       // self loop included
        dinv[i]   = rsqrtf(d);
        invdeg[i] = 1.0f / d;
    }
}

// ---------------------------------------------------------------------------
// WMMA GEMM: out[N,128] = A[N,128] @ W[128,128]
// One block = 16 rows x 128 cols; 8 waves, each wave owns one 16x16 tile.
// K=128 swept in 32 steps of v_wmma_f32_16x16x4_f32 (fp32-exact matrix path).
// ---------------------------------------------------------------------------
__global__ __launch_bounds__(256) void gemm128_wmma(const float* __restrict__ A,
                                                    const float* __restrict__ W,
                                                    float* __restrict__ out, int N) {
    __shared__ float sW[HDIM * HDIM];      // 64 KB  (<< 320 KB/WGP)
    __shared__ float sA[16 * 129];         // padded stride: no 16-way bank conflict

    const int tid = threadIdx.x;
    const int row0 = blockIdx.x * 16;

    for (int i = tid; i < HDIM * HDIM; i += 256) sW[i] = W[i];
    for (int i = tid; i < 16 * HDIM; i += 256) {
        int r = i >> 7, c = i & 127;
        int gr = row0 + r;
        sA[r * 129 + c] = (gr < N) ? A[(long long)gr * HDIM + c] : 0.0f;
    }
    __syncthreads();

    const int wave = tid >> 5;             // 0..7 -> 16-col tile
    const int lane = tid & 31;
    const int n0   = wave * 16;
    const int mrow = lane & 15;            // M index within tile
    const int khalf = (lane >> 4) * 2;     // lanes 16-31 carry K+2,K+3

    v8f acc = {};
    #pragma unroll 4
    for (int k0 = 0; k0 < HDIM; k0 += 4) {
        v2f a, b;
        // A 16x4 frag: lane<16 -> A[m][k0], A[m][k0+1]; lane>=16 -> +2,+3
        a.x = sA[mrow * 129 + k0 + khalf];
        a.y = sA[mrow * 129 + k0 + khalf + 1];
        // B 4x16 frag: row K striped across lanes within a VGPR
        b.x = sW[(k0 + khalf) * HDIM + n0 + mrow];
        b.y = sW[(k0 + khalf + 1) * HDIM + n0 + mrow];
        acc = __builtin_amdgcn_wmma_f32_16x16x4_f32(false, a, false, b,
                                                    (short)0, acc, false, false);
    }

    // C/D layout: VGPR r -> lanes 0-15: M=r ; lanes 16-31: M=r+8 ; N = lane&15
    // Rows within a tile are a constant 128-float stride apart: hoist the bound
    // check to one test per wave and emit 8 unpredicated stores with immediate
    // offsets (kills the per-row exec-mask/branch sequence).
    const int col   = n0 + mrow;
    const int rbase = (lane >> 4) * 8;
    float* op = out + (long long)(row0 + rbase) * HDIM + col;
    if (row0 + 16 <= N) {                  // full tile (always true for N%16==0)
        #pragma unroll
        for (int r = 0; r < 8; ++r) op[r * HDIM] = acc[r];
    } else {
        #pragma unroll
        for (int r = 0; r < 8; ++r)
            if (row0 + rbase + r < N) op[r * HDIM] = acc[r];
    }
}

// ---------------------------------------------------------------------------
// edge scatter: agg[dst] += xw[src] * dinv[src]*dinv[dst]  (one wave / edge)
// src/dst/norm are wave-uniform -> force scalar via readfirstlane.
// ---------------------------------------------------------------------------
__global__ __launch_bounds__(256) void scatter_kernel(const int* __restrict__ src,
                                                      const int* __restrict__ dst,
                                                      const float* __restrict__ dinv,
                                                      const float* __restrict__ xw,
                                                      float* __restrict__ agg, int E) {
    int lane = threadIdx.x & 31;
    int wid  = (blockIdx.x * blockDim.x + threadIdx.x) >> 5;
    if (wid >= E) return;
    int s = __builtin_amdgcn_readfirstlane(src[wid]);
    int d = __builtin_amdgcn_readfirstlane(dst[wid]);
    // hint next edge batch into cache (global_prefetch_b8)
    if (wid + 8 < E) __builtin_prefetch(&src[wid + 8], 0, 1);
    float norm = dinv[s] * dinv[d];
    const float* xs = xw + (long long)s * HDIM;
    float*       ad = agg + (long long)d * HDIM;
    #pragma unroll
    for (int j = 0; j < 4; ++j) {
        int c = lane + j * 32;
        atomicAdd(&ad[c], xs[c] * norm);
    }
}

// ---------------------------------------------------------------------------
// finalize: h = relu(agg + xw * invdeg[row] + b[col])
// ---------------------------------------------------------------------------
__global__ void finalize_kernel(const float* __restrict__ agg,
                                const float* __restrict__ xw,
                                const float* __restrict__ invdeg,
                                const float* __restrict__ b,
                                float* __restrict__ h, int N) {
    long long t = (long long)blockIdx.x * blockDim.x + threadIdx.x;
    if (t >= (long long)N * HDIM) return;
    int row = (int)(t >> 7), col = (int)(t & 127);
    float v = agg[t] + xw[t] * invdeg[row] + b[col];
    h[t] = fmaxf(v, 0.0f);
}

// ---------------------------------------------------------------------------
// pooling: pooled[g] += h[i] ; cnt[g] += 1
// ---------------------------------------------------------------------------
__global__ void pool_sum_kernel(const float* __restrict__ h, const int* __restrict__ batch,
                                float* __restrict__ pooled, int N) {
    long long t = (long long)blockIdx.x * blockDim.x + threadIdx.x;
    if (t >= (long long)N * HDIM) return;
    int node = (int)(t >> 7), col = (int)(t & 127);
    atomicAdd(&pooled[(long long)batch[node] * HDIM + col], h[t]);
}

__global__ void pool_cnt_kernel(const int* __restrict__ batch, float* __restrict__ cnt, int N) {
    int i = blockIdx.x * blockDim.x + threadIdx.x;
    if (i < N) atomicAdd(&cnt[batch[i]], 1.0f);
}

// ---------------------------------------------------------------------------
// classifier: out[64,10] = (pooled/cnt) @ Wc[128,10] + bc
// ---------------------------------------------------------------------------
__global__ void classify_kernel(const float* __restrict__ pooled, const float* __restrict__ cnt,
                                const float* __restrict__ Wc, const float* __restrict__ bc,
                                float* __restrict__ out) {
    int t = threadIdx.x;                   // 640 threads: g=t/10, c=t%10
    if (t >= NGRAPH * 10) return;
    int g = t / 10, c = t % 10;
    float inv = 1.0f / fmaxf(cnt[g], 1.0f);
    float acc = bc[c];
    #pragma unroll 8
    for (int k = 0; k < HDIM; ++k)
        acc += pooled[g * HDIM + k] * inv * Wc[k * 10 + c];
    out[g * 10 + c] = acc;
}

// ---------------------------------------------------------------------------
extern "C" void kernel_launch(void* const* d_in, const int* in_sizes, int n_in,
                              void* d_out, int out_size, void* d_ws, size_t ws_size,
                              hipStream_t stream) {
    const float* x     = (const float*)d_in[0];
    const int*   ei    = (const int*)  d_in[1];
    const int*   batch = (const int*)  d_in[2];
    const float* W1 = (const float*)d_in[3];
    const float* b1 = (const float*)d_in[4];
    const float* W2 = (const float*)d_in[5];
    const float* b2 = (const float*)d_in[6];
    const float* W3 = (const float*)d_in[7];
    const float* b3 = (const float*)d_in[8];
    const float* Wc = (const float*)d_in[9];
    const float* bc = (const float*)d_in[10];

    const int N = in_sizes[0] / HDIM;      // 50000
    const int E = in_sizes[1] / 2;         // 800000
    const int* src = ei;
    const int* dst = ei + E;

    const long long NH = (long long)N * HDIM;
    float* ws     = (float*)d_ws;
    float* xw     = ws;                    // [N,128]
    float* agg    = ws + NH;               // [N,128]
    float* ha     = ws + 2 * NH;           // [N,128]
    float* hb     = ws + 3 * NH;           // [N,128]
    float* deg    = ws + 4 * NH;           // [N]
    float* dinv   = deg + N;               // [N]
    float* invdeg = deg + 2 * (long long)N;// [N]
    float* pooled = deg + 3 * (long long)N;// [64,128]
    float* cnt    = pooled + NGRAPH * HDIM;// [64]

    const int T = 256;
    auto blocks = [](long long n, int t) { return (unsigned)((n + t - 1) / t); };

    // degree / norms (recomputed each call: deterministic)
    zero_kernel<<<blocks(N, T), T, 0, stream>>>(deg, N);
    deg_kernel<<<blocks(E, T), T, 0, stream>>>(dst, deg, E);
    dinv_kernel<<<blocks(N, T), T, 0, stream>>>(deg, dinv, invdeg, N);

    const float* hin = x;
    const float* Ws[3] = {W1, W2, W3};
    const float* bs[3] = {b1, b2, b3};
    float* houts[3] = {ha, hb, ha};

    for (int L = 0; L < 3; ++L) {
        gemm128_wmma<<<blocks((long long)N, 16), T, 0, stream>>>(hin, Ws[L], xw, N);
        zero_kernel<<<blocks(NH, T), T, 0, stream>>>(agg, NH);
        scatter_kernel<<<blocks((long long)E * 32, T), T, 0, stream>>>(src, dst, dinv, xw, agg, E);
        finalize_kernel<<<blocks(NH, T), T, 0, stream>>>(agg, xw, invdeg, bs[L], houts[L], N);
        hin = houts[L];
    }

    // global mean pool + classifier
    zero_kernel<<<blocks(NGRAPH * HDIM + NGRAPH, T), T, 0, stream>>>(pooled, NGRAPH * HDIM + NGRAPH);
    pool_sum_kernel<<<blocks(NH, T), T, 0, stream>>>(hin, batch, pooled, N);
    pool_cnt_kernel<<<blocks(N, T), T, 0, stream>>>(batch, cnt, N);
    classify_kernel<<<1, 640, 0, stream>>>(pooled, cnt, Wc, bc, (float*)d_out);
}